// SpectralConv1dODE_x_5669356830907
// MI455X (gfx1250) — compile-verified
//
#include <hip/hip_runtime.h>

// MI455X / gfx1250, wave32. fp32 WMMA path: V_WMMA_F32_16X16X4_F32.
// D(16x16,f32) = A(16x4,f32) * B(4x16,f32) + C
// Lane layout (wave32): lo = lane&15, hi = lane>>4
//   A frag: a[i] = A[row=lo][k = 4*kk + 2*hi + i]
//   B frag: b[i] = B[k = 4*kk + 2*hi + i][col=lo]
//   C/D:    c[i] = C[row = 8*hi + i][col = lo]

typedef __attribute__((ext_vector_type(2))) float v2f;
typedef __attribute__((ext_vector_type(8))) float v8f;

__device__ __forceinline__ v8f wmma4(v2f a, v2f b, v8f c) {
  return __builtin_amdgcn_wmma_f32_16x16x4_f32(false, a, false, b, (short)0, c,
                                               false, false);
}

// LDS byte offset of a __shared__ object: flat LDS aperture keeps the
// wave-relative LDS offset in addr[31:0] (ISA 10.2 aperture mapping).
__device__ __forceinline__ unsigned lds_offset(const void* p) {
  return (unsigned)(size_t)p;
}

// Async global->LDS copy, 4B per lane, tracked by ASYNCcnt (CDNA5 TDM-class
// data movement; inline asm so it works on both toolchains).
__device__ __forceinline__ void async_copy_b32(unsigned lds_off,
                                               const void* gptr) {
  asm volatile("global_load_async_to_lds_b32 %0, %1, off" ::"v"(lds_off),
               "v"((unsigned long long)(size_t)gptr)
               : "memory");
}
__device__ __forceinline__ void wait_async0() {
  asm volatile("s_wait_asynccnt 0" ::: "memory");
}

// B=8, N=1024, E=8, C=128, M=128, T=10, SUBSTEPS=2
// ys workspace layout: [T][M][BE=64][C=128] fp32 (40 MB, L2-resident)

// ---------------------------------------------------------------------------
// Kernel 1: xhat0[m, b, e, c] = sum_n U[b][n][m] * x[b][n][e*128+c]
// b = blockIdx.x, e-column-block = blockIdx.y. Output 128(m) x 128(ec) per
// block; wave w owns m-tile w and all 8 ec tiles. A-frag loads from U and
// B-frag loads from x are coalesced 64B groups; no LDS needed.
// ---------------------------------------------------------------------------
__global__ __launch_bounds__(256) void spectral_project_in(
    const float* __restrict__ U, const float* __restrict__ x,
    float* __restrict__ ys0) {
  const int b = blockIdx.x;
  const int e = blockIdx.y;
  const int ec0 = e * 128;
  const int tid = threadIdx.x;
  const int lane = tid & 31;
  const int w = tid >> 5;
  const int lo = lane & 15;
  const int hi = lane >> 4;
  const float* __restrict__ Ub = U + (size_t)b * 1024 * 1024;
  const float* __restrict__ xb = x + (size_t)b * 1024 * 1024;

  v8f acc[8];
#pragma unroll
  for (int j = 0; j < 8; ++j)
#pragma unroll
    for (int i = 0; i < 8; ++i) acc[j][i] = 0.0f;

  const int mcol = w * 16 + lo;
  for (int kk = 0; kk < 256; ++kk) {  // K = 1024, 4 per WMMA
    const int kb = kk * 4 + hi * 2;
    v2f a;
    a.x = Ub[(size_t)kb * 1024 + mcol];
    a.y = Ub[(size_t)(kb + 1) * 1024 + mcol];
#pragma unroll
    for (int j = 0; j < 8; ++j) {
      const int col = ec0 + j * 16 + lo;
      v2f bb;
      bb.x = xb[(size_t)kb * 1024 + col];
      bb.y = xb[(size_t)(kb + 1) * 1024 + col];
      acc[j] = wmma4(a, bb, acc[j]);
    }
  }

  // ys0[(m*64 + b*8 + e)*128 + c]
#pragma unroll
  for (int j = 0; j < 8; ++j) {
#pragma unroll
    for (int i = 0; i < 8; ++i) {
      const int m = w * 16 + hi * 8 + i;
      const int c = j * 16 + lo;
      ys0[((size_t)m * 64 + b * 8 + e) * 128 + c] = acc[j][i];
    }
  }
}

// ---------------------------------------------------------------------------
// Kernel 2: RK4 time integration, one workgroup per mode m (128 blocks).
// y[m] (64x128) and the RK accumulators live in VGPRs for all 18 substeps;
// stage inputs are exchanged via padded LDS t1[c][be]; W[m] is staged once
// into padded LDS Wl (100 KB total LDS -- CDNA5 allows 320 KB/WG).
// Padding: stride 132/68 floats so the hi=1 half-wave's K-rows (+2 rows)
// land 8/+ banks away from the hi=0 half (64-bank LDS, conflict-free).
// ---------------------------------------------------------------------------
__global__ __launch_bounds__(256) void spectral_rk4(
    const float* __restrict__ W, const float* __restrict__ times,
    float* __restrict__ ys) {
  const int m = blockIdx.x;
  __shared__ float Wl[128][132];  // [i][o]
  __shared__ float t1[128][68];   // [c][be]
  const int tid = threadIdx.x;
  const int lane = tid & 31;
  const int w = tid >> 5;
  const int lo = lane & 15;
  const int hi = lane >> 4;
  const int r = w & 3;          // be row-tile (0..3)
  const int cg = (w >> 2) * 4;  // first c col-tile (0 or 4)
  const float* __restrict__ Wm = W + (size_t)m * 128 * 128;

  // stage W[m] into LDS (coalesced global, conflict-free LDS writes)
  for (int idx = tid; idx < 128 * 128; idx += 256)
    Wl[idx >> 7][idx & 127] = Wm[idx];

  v8f y[4], acc[4], kst[4];
  const float* __restrict__ ym0 = ys + (size_t)m * 64 * 128;
#pragma unroll
  for (int j = 0; j < 4; ++j) {
    const int cj = cg + j;
#pragma unroll
    for (int i = 0; i < 8; ++i) {
      const int be = r * 16 + hi * 8 + i;
      y[j][i] = ym0[be * 128 + cj * 16 + lo];
    }
  }

  auto write_stage = [&](float ch, bool use_k) {
#pragma unroll
    for (int j = 0; j < 4; ++j) {
      const int cj = cg + j;
#pragma unroll
      for (int i = 0; i < 8; ++i) {
        const int be = r * 16 + hi * 8 + i;
        const int c = cj * 16 + lo;
        t1[c][be] = use_k ? (y[j][i] + ch * kst[j][i]) : y[j][i];
      }
    }
  };

  auto gemm_stage = [&]() {
#pragma unroll
    for (int j = 0; j < 4; ++j)
#pragma unroll
      for (int i = 0; i < 8; ++i) kst[j][i] = 0.0f;
    for (int kk = 0; kk < 32; ++kk) {  // K = C = 128
      const int kb = kk * 4 + hi * 2;
      v2f a;
      a.x = t1[kb][r * 16 + lo];
      a.y = t1[kb + 1][r * 16 + lo];
#pragma unroll
      for (int j = 0; j < 4; ++j) {
        const int cj = cg + j;
        v2f bb;
        bb.x = Wl[kb][cj * 16 + lo];
        bb.y = Wl[kb + 1][cj * 16 + lo];
        kst[j] = wmma4(a, bb, kst[j]);
      }
    }
  };

  for (int it = 0; it < 9; ++it) {
    const float dt = times[it + 1] - times[it];
    const float h = 0.5f * dt;  // SUBSTEPS = 2
    for (int ss = 0; ss < 2; ++ss) {
      // k1
      __syncthreads();
      write_stage(0.0f, false);
      __syncthreads();
      gemm_stage();
#pragma unroll
      for (int j = 0; j < 4; ++j)
#pragma unroll
        for (int i = 0; i < 8; ++i) acc[j][i] = kst[j][i];
      // k2
      __syncthreads();
      write_stage(0.5f * h, true);
      __syncthreads();
      gemm_stage();
#pragma unroll
      for (int j = 0; j < 4; ++j)
#pragma unroll
        for (int i = 0; i < 8; ++i) acc[j][i] += 2.0f * kst[j][i];
      // k3
      __syncthreads();
      write_stage(0.5f * h, true);
      __syncthreads();
      gemm_stage();
#pragma unroll
      for (int j = 0; j < 4; ++j)
#pragma unroll
        for (int i = 0; i < 8; ++i) acc[j][i] += 2.0f * kst[j][i];
      // k4
      __syncthreads();
      write_stage(h, true);
      __syncthreads();
      gemm_stage();
      const float h6 = h * (1.0f / 6.0f);
#pragma unroll
      for (int j = 0; j < 4; ++j)
#pragma unroll
        for (int i = 0; i < 8; ++i) y[j][i] += h6 * (acc[j][i] + kst[j][i]);
    }
    // snapshot ys[it+1][m]
    float* __restrict__ yt = ys + ((size_t)(it + 1) * 128 + m) * 64 * 128;
#pragma unroll
    for (int j = 0; j < 4; ++j) {
      const int cj = cg + j;
#pragma unroll
      for (int i = 0; i < 8; ++i) {
        const int be = r * 16 + hi * 8 + i;
        yt[be * 128 + cj * 16 + lo] = y[j][i];
      }
    }
  }
}

// ---------------------------------------------------------------------------
// Kernel 3: x_out[b,t,n,e,c] = sum_m U[b][n][m] * ys[t][m][b*8+e][c]
// Grid: (n-tile 16, e 8, t*8+b 80). Block output = 64(n) x 128(c), K = M=128.
// U is staged transposed into padded LDS via async global->LDS b32 copies
// (ASYNCcnt path). B-frags from ys read directly (coalesced, L2 resident).
// Output (335 MB, write-once) uses non-temporal stores so it does not evict
// U/ys from the 192 MB L2.
// ---------------------------------------------------------------------------
__global__ __launch_bounds__(256) void spectral_project_out(
    const float* __restrict__ U, const float* __restrict__ ys,
    float* __restrict__ out) {
  const int n0 = blockIdx.x * 64;
  const int e = blockIdx.y;
  const int t = blockIdx.z >> 3;
  const int b = blockIdx.z & 7;
  __shared__ float As[128][68];  // [m][nn], padded
  const int tid = threadIdx.x;
  const int lane = tid & 31;
  const int w = tid >> 5;
  const int lo = lane & 15;
  const int hi = lane >> 4;
  const int r = w & 3;          // n row-tile (0..3)
  const int cg = (w >> 2) * 4;  // first c col-tile

  const float* __restrict__ Ub = U + (size_t)b * 1024 * 1024;
  // ys[t][m][b*8+e][c] = yt[m*8192 + c]
  const float* __restrict__ yt =
      ys + ((size_t)t * 128 * 64 + (size_t)(b * 8 + e)) * 128;

  // async transpose-stage: global reads coalesced in m (fast index),
  // LDS writes land at As[m][n] with conflict-breaking row pad.
  for (int idx = tid; idx < 128 * 64; idx += 256) {
    const int mm = idx & 127;
    const int nn = idx >> 7;
    async_copy_b32(lds_offset(&As[mm][nn]),
                   Ub + (size_t)(n0 + nn) * 1024 + mm);
  }
  wait_async0();
  __syncthreads();

  v8f acc[4];
#pragma unroll
  for (int j = 0; j < 4; ++j)
#pragma unroll
    for (int i = 0; i < 8; ++i) acc[j][i] = 0.0f;

  for (int kk = 0; kk < 32; ++kk) {  // K = M = 128
    const int kb = kk * 4 + hi * 2;
    v2f a;
    a.x = As[kb][r * 16 + lo];
    a.y = As[kb + 1][r * 16 + lo];
#pragma unroll
    for (int j = 0; j < 4; ++j) {
      const int cj = cg + j;
      v2f bb;
      bb.x = yt[(size_t)kb * 64 * 128 + cj * 16 + lo];
      bb.y = yt[(size_t)(kb + 1) * 64 * 128 + cj * 16 + lo];
      acc[j] = wmma4(a, bb, acc[j]);
    }
  }

  // out[(((b*10 + t)*1024 + n)*8 + e)*128 + c], streaming NT stores
  float* __restrict__ ob = out + (size_t)(b * 10 + t) * 1024 * 1024;
#pragma unroll
  for (int j = 0; j < 4; ++j) {
    const int cj = cg + j;
#pragma unroll
    for (int i = 0; i < 8; ++i) {
      const int n = n0 + r * 16 + hi * 8 + i;
      __builtin_nontemporal_store(
          acc[j][i], &ob[(size_t)n * 1024 + e * 128 + cj * 16 + lo]);
    }
  }
}

extern "C" void kernel_launch(void* const* d_in, const int* in_sizes, int n_in,
                              void* d_out, int out_size, void* d_ws,
                              size_t ws_size, hipStream_t stream) {
  (void)in_sizes; (void)n_in; (void)out_size; (void)ws_size;
  const float* x = (const float*)d_in[0];      // (8,1024,8,128)
  const float* U = (const float*)d_in[1];      // (8,1024,1024)
  const float* W = (const float*)d_in[2];      // (128,128,128)
  const float* times = (const float*)d_in[3];  // (10,)
  float* out = (float*)d_out;                  // (8,10,1024,8,128)
  float* ys = (float*)d_ws;                    // [10][128][64][128] f32

  spectral_project_in<<<dim3(8, 8), 256, 0, stream>>>(U, x, ys);
  spectral_rk4<<<dim3(128), 256, 0, stream>>>(W, times, ys);
  spectral_project_out<<<dim3(16, 8, 80), 256, 0, stream>>>(U, ys, out);
}